// SIADecoder_86328842650011
// MI455X (gfx1250) — compile-verified
//
#include <hip/hip_runtime.h>
#include <math.h>
#include <stdint.h>

typedef __attribute__((ext_vector_type(2))) float v2f;
typedef __attribute__((ext_vector_type(8))) float v8f;

#define B_ 8
#define S_ 2048
#define D_ 512
#define NH 4
#define NB 32
#define BS 64
#define CHUNKS 128          // NH * NB per batch
#define HS (NH * S_)        // 8192
#define SCALE_D 0.044194173824159216f   // 512^-0.5
#define NEG_BIG -1.7014118e38f          // -FLT_MAX/2
#define SELF_VAL -5.0e4f

__device__ __forceinline__ v8f wmma4(v2f a, v2f b, v8f c) {
  // V_WMMA_F32_16X16X4_F32 : D = A(16x4) x B(4x16) + C(16x16), f32
  return __builtin_amdgcn_wmma_f32_16x16x4_f32(false, a, false, b, (short)0, c,
                                               false, false);
}

// Async copy 16B global -> LDS (per active lane), tracked by ASYNCcnt.
// GV mode: 64-bit address in VGPR pair, SADDR=off. LDS byte address is the
// low 32 bits of the generic pointer (flat->LDS aperture truncation).
__device__ __forceinline__ void async_b128(void* lds, const void* gaddr) {
  asm volatile("global_load_async_to_lds_b128 %0, %1, off"
               :: "v"((unsigned)(uintptr_t)lds), "v"(gaddr)
               : "memory");
}
__device__ __forceinline__ void wait_async0() {
  asm volatile("s_wait_asynccnt 0x0" ::: "memory");
}

// ---------------------------------------------------------------- hashing ----
__global__ __launch_bounds__(256) void lsh_hash_kernel(
    const float* __restrict__ qk, const float* __restrict__ rot,
    unsigned int* __restrict__ keys) {
  __shared__ float rbuf[8][64];
  const int wave = threadIdx.x >> 5, lane = threadIdx.x & 31;
  const int token = blockIdx.x * 8 + wave;          // [0, B*S)
  const int b = token >> 11, s = token & (S_ - 1);
  const float* q = qk + (size_t)token * D_;
  float acc0 = 0.f, acc1 = 0.f;                     // outputs o=lane, o=lane+32
  for (int d0 = 0; d0 < D_; d0 += 32) {
    float qv = q[d0 + lane];
    for (int j = 0; j < 32; ++j) {
      float x = __shfl(qv, j, 32);
      const float* rp = rot + (size_t)(d0 + j) * 64;  // rot[d][h*16+i]
      acc0 = fmaf(x, rp[lane], acc0);
      acc1 = fmaf(x, rp[lane + 32], acc1);
    }
  }
  rbuf[wave][lane] = acc0;
  rbuf[wave][lane + 32] = acc1;
  __syncthreads();
  if (lane < NH) {
    const int h = lane;
    // argmax over concat([r, -r]) with first-index tiebreak (jnp.argmax)
    float best = rbuf[wave][h * 16];
    int bi = 0;
    for (int idx = 1; idx < 32; ++idx) {
      float v = (idx < 16) ? rbuf[wave][h * 16 + idx]
                           : -rbuf[wave][h * 16 + idx - 16];
      if (v > best) { best = v; bi = idx; }
    }
    unsigned key = (unsigned)S_ * (unsigned)(bi + h * NB) + (unsigned)s;
    keys[((size_t)b * NH + h) * S_ + s] = key;
  }
}

// ------------------------------------------------------------------- sort ----
__global__ __launch_bounds__(1024) void lsh_sort_kernel(
    const unsigned int* __restrict__ keys, unsigned int* __restrict__ stp) {
  __shared__ unsigned long long arr[HS];            // 64 KB
  const int b = blockIdx.x;
  for (int t = threadIdx.x; t < HS; t += 1024)
    arr[t] = ((unsigned long long)keys[(size_t)b * HS + t] << 32) | (unsigned)t;
  for (int k = 2; k <= HS; k <<= 1) {
    for (int j = k >> 1; j > 0; j >>= 1) {
      __syncthreads();
      for (int t = threadIdx.x; t < HS; t += 1024) {
        int ixj = t ^ j;
        if (ixj > t) {
          unsigned long long a = arr[t], c = arr[ixj];
          bool up = ((t & k) == 0);
          if ((a > c) == up) { arr[t] = c; arr[ixj] = a; }
        }
      }
    }
  }
  __syncthreads();
  for (int t = threadIdx.x; t < HS; t += 1024) {
    unsigned r = (unsigned)arr[t];                  // original index h*S + s
    unsigned s = r & (S_ - 1), h = r >> 11;
    stp[(size_t)b * HS + t] = s | (h << 16);        // packed position
  }
}

// -------------------------------------------------------- chunk attention ----
__global__ __launch_bounds__(256) void lsh_attn_kernel(
    const float* __restrict__ qk, const float* __restrict__ v,
    const unsigned int* __restrict__ stp, const int* __restrict__ mask,
    float* __restrict__ o4, float* __restrict__ sl4) {
  __shared__ float sdots[64][130];                  // 33280 B (dots -> probs)
  __shared__ float sbuf[2][128][24];                // 24576 B async stage
                                                    //  (rows 16B aligned)
  __shared__ unsigned int spos[128];
  __shared__ float sinv[128];
  __shared__ int smask[128];

  const int b = blockIdx.x >> 7;
  const int c = blockIdx.x & (CHUNKS - 1);
  const int cp = (c + CHUNKS - 1) & (CHUNKS - 1);   // look-one-back chunk
  const int tid = threadIdx.x, lane = tid & 31, wave = tid >> 5;
  const int half = lane >> 4, l16 = lane & 15, kb = half * 2;

  if (tid < 128) {
    int p = (tid < BS) ? (c * BS + tid) : (cp * BS + tid - BS);
    unsigned pk = stp[(size_t)b * HS + p];
    spos[tid] = pk;
    smask[tid] = mask[(size_t)b * S_ + (pk & 0xFFFFu)];
  }
  __syncthreads();

  // key-row inverse norms:  bk = row / max(||row||, 1e-12)
  for (int j = wave; j < 128; j += 8) {
    const float* r = qk + ((size_t)b * S_ + (spos[j] & 0xFFFFu)) * D_;
    float ss = 0.f;
    for (int t = lane; t < D_; t += 32) { float x = r[t]; ss = fmaf(x, x, ss); }
    for (int o = 16; o > 0; o >>= 1) ss += __shfl_xor(ss, o, 32);
    if (lane == 0) sinv[j] = 1.f / fmaxf(sqrtf(ss), 1e-12f);
  }
  __syncthreads();

  // ---- dots = bq @ bk^T : K streamed in 16-wide slices, double-buffered
  //      via GLOBAL_LOAD_ASYNC_TO_LDS_B128, f32 WMMA ----
  v8f acc[4];
  for (int t = 0; t < 4; ++t) acc[t] = (v8f){0.f,0.f,0.f,0.f,0.f,0.f,0.f,0.f};
  const int tn = wave;                              // N tile 0..7 (cols 0..127)

  for (int t = tid; t < 512; t += 256) {            // prime slice 0
    int row = t >> 2, q4 = t & 3;
    async_b128(&sbuf[0][row][q4 * 4],
               qk + ((size_t)b * S_ + (spos[row] & 0xFFFFu)) * D_ + q4 * 4);
  }
  for (int sl = 0; sl < D_ / 16; ++sl) {
    const int cur = sl & 1;
    wait_async0();                                  // my copies (slice sl) done
    __syncthreads();                                // everyone's done; prev
                                                    // slice fully consumed
    if (sl + 1 < D_ / 16) {                         // prefetch next slice
      for (int t = tid; t < 512; t += 256) {
        int row = t >> 2, q4 = t & 3;
        async_b128(&sbuf[cur ^ 1][row][q4 * 4],
                   qk + ((size_t)b * S_ + (spos[row] & 0xFFFFu)) * D_ +
                       (sl + 1) * 16 + q4 * 4);
      }
    }
    for (int tm = 0; tm < 4; ++tm) {                // M tiles (query rows 0..63)
      for (int kk = 0; kk < 16; kk += 4) {
        v2f a = *(const v2f*)&sbuf[cur][tm * 16 + l16][kk + kb];  // A[m][k..k+1]
        v2f bb = *(const v2f*)&sbuf[cur][tn * 16 + l16][kk + kb]; // B[k][n]
        acc[tm] = wmma4(a, bb, acc[tm]);
      }
    }
  }
  // scale, mask, store dots
  {
    const int j = tn * 16 + l16;
    const unsigned sj = spos[j] & 0xFFFFu;
    const float colscale = SCALE_D * sinv[j];
    const int mj = smask[j];
    for (int tm = 0; tm < 4; ++tm) {
      for (int r = 0; r < 8; ++r) {
        int i = tm * 16 + half * 8 + r;
        float d = acc[tm][r] * colscale;
        if (!(smask[i] & mj)) d = NEG_BIG;               // input mask
        if ((spos[i] & 0xFFFFu) == sj) d = SELF_VAL;     // self mask overwrites
        sdots[i][j] = d;
      }
    }
  }
  __syncthreads();

  // ---- row softmax + logsumexp ----
  for (int i = wave; i < 64; i += 8) {
    float mx = -3.4e38f;
    for (int j = lane; j < 128; j += 32) mx = fmaxf(mx, sdots[i][j]);
    for (int o = 16; o > 0; o >>= 1) mx = fmaxf(mx, __shfl_xor(mx, o, 32));
    float sm = 0.f;
    for (int j = lane; j < 128; j += 32) sm += expf(sdots[i][j] - mx);
    for (int o = 16; o > 0; o >>= 1) sm += __shfl_xor(sm, o, 32);
    float lse = mx + logf(sm);
    for (int j = lane; j < 128; j += 32) sdots[i][j] = expf(sdots[i][j] - lse);
    if (lane == 0) {
      unsigned pk = spos[i];
      sl4[((size_t)b * NH + (pk >> 16)) * S_ + (pk & 0xFFFFu)] = lse;
    }
  }

  // ---- bo = probs @ bv : N streamed in 32-wide slices (async staged) ----
  float (*vstage)[36] = (float (*)[36]) & sbuf[0][0][0];   // 18432 B view
  const int tm2 = wave & 3, tnl = wave >> 2;        // 8 waves -> 4x2 tiles
  for (int ns = 0; ns < D_ / 32; ++ns) {
    __syncthreads();                                // prev slice reads done
    for (int t = tid; t < 1024; t += 256) {         // 128 rows x 8 float4
      int row = t >> 3, q4 = t & 7;
      async_b128(&vstage[row][q4 * 4],
                 v + ((size_t)b * S_ + (spos[row] & 0xFFFFu)) * D_ + ns * 32 +
                     q4 * 4);
    }
    wait_async0();
    __syncthreads();
    v8f o = (v8f){0.f,0.f,0.f,0.f,0.f,0.f,0.f,0.f};
    for (int kk = 0; kk < 128; kk += 4) {
      v2f a = *(const v2f*)&sdots[tm2 * 16 + l16][kk + kb];     // probs[m][k]
      v2f bb;
      bb[0] = vstage[kk + kb][tnl * 16 + l16];                  // bv[k][n]
      bb[1] = vstage[kk + kb + 1][tnl * 16 + l16];
      o = wmma4(a, bb, o);
    }
    const int col = ns * 32 + tnl * 16 + l16;
    for (int r = 0; r < 8; ++r) {
      int i = tm2 * 16 + half * 8 + r;
      unsigned pk = spos[i];
      o4[(((size_t)b * NH + (pk >> 16)) * S_ + (pk & 0xFFFFu)) * D_ + col] =
          o[r];
    }
  }
}

// ----------------------------------------- hash-round combine + layernorm ----
__global__ __launch_bounds__(256) void combine_ln_kernel(
    const float* __restrict__ o4, const float* __restrict__ sl4,
    const float* __restrict__ lnw, const float* __restrict__ lnb,
    float* __restrict__ out) {
  const int token = blockIdx.x;
  const int b = token >> 11, s = token & (S_ - 1);
  __shared__ float wts[4];
  __shared__ float red[2][8];
  const int tid = threadIdx.x, lane = tid & 31, wave = tid >> 5;
  if (tid == 0) {
    float l[4], mx = -3.4e38f;
    for (int h = 0; h < 4; ++h) {
      l[h] = sl4[((size_t)b * NH + h) * S_ + s];
      mx = fmaxf(mx, l[h]);
    }
    float sm = 0.f;
    for (int h = 0; h < 4; ++h) { l[h] = expf(l[h] - mx); sm += l[h]; }
    float inv = 1.f / sm;
    for (int h = 0; h < 4; ++h) wts[h] = l[h] * inv;
  }
  __syncthreads();
  float vals[2], s1 = 0.f, s2 = 0.f;
  for (int u = 0; u < 2; ++u) {
    int col = tid + u * 256;
    float a = 0.f;
    for (int h = 0; h < 4; ++h)
      a = fmaf(wts[h], o4[(((size_t)b * NH + h) * S_ + s) * D_ + col], a);
    vals[u] = a; s1 += a; s2 = fmaf(a, a, s2);
  }
  for (int o = 16; o > 0; o >>= 1) {
    s1 += __shfl_xor(s1, o, 32);
    s2 += __shfl_xor(s2, o, 32);
  }
  if (lane == 0) { red[0][wave] = s1; red[1][wave] = s2; }
  __syncthreads();
  if (tid == 0) {
    float a = 0.f, c = 0.f;
    for (int w = 0; w < 8; ++w) { a += red[0][w]; c += red[1][w]; }
    red[0][0] = a; red[1][0] = c;
  }
  __syncthreads();
  const float mu = red[0][0] * (1.f / 512.f);
  const float var = red[1][0] * (1.f / 512.f) - mu * mu;
  const float rs = rsqrtf(var + 1e-3f);
  for (int u = 0; u < 2; ++u) {
    int col = tid + u * 256;
    out[(size_t)token * D_ + col] = (vals[u] - mu) * rs * lnw[col] + lnb[col];
  }
}

// ------------------------------------------------------------------ launch ---
extern "C" void kernel_launch(void* const* d_in, const int* in_sizes, int n_in,
                              void* d_out, int out_size, void* d_ws,
                              size_t ws_size, hipStream_t stream) {
  const float* x = (const float*)d_in[0];
  const float* y = (const float*)d_in[1];
  // d_in[2] = source_mask (unused by reference)
  const int* tgt_mask = (const int*)d_in[3];
  const float* rot1 = (const float*)d_in[4];
  const float* rot2 = (const float*)d_in[5];
  const float* lnw = (const float*)d_in[6];
  const float* lnb = (const float*)d_in[7];
  float* out = (float*)d_out;

  char* ws = (char*)d_ws;
  unsigned int* keys = (unsigned int*)ws;  ws += (size_t)B_ * HS * 4;
  unsigned int* stp = (unsigned int*)ws;   ws += (size_t)B_ * HS * 4;
  float* sl4 = (float*)ws;                 ws += (size_t)B_ * HS * 4;
  float* o4 = (float*)ws;                  ws += (size_t)B_ * HS * D_ * 4;
  float* tgt = (float*)ws;                 // B*S*D floats

  const size_t BSD = (size_t)B_ * S_ * D_;

  // pass 1: tgt = LN(lsh_attn(qk=y, v=y, tgt_mask, rot1))
  lsh_hash_kernel<<<(B_ * S_) / 8, 256, 0, stream>>>(y, rot1, keys);
  lsh_sort_kernel<<<B_, 1024, 0, stream>>>(keys, stp);
  lsh_attn_kernel<<<B_ * CHUNKS, 256, 0, stream>>>(y, y, stp, tgt_mask, o4, sl4);
  combine_ln_kernel<<<B_ * S_, 256, 0, stream>>>(o4, sl4, lnw, lnb, tgt);

  // pass 2: x_out = LN(lsh_attn(qk=tgt, v=x, tgt_mask, rot2))
  lsh_hash_kernel<<<(B_ * S_) / 8, 256, 0, stream>>>(tgt, rot2, keys);
  lsh_sort_kernel<<<B_, 1024, 0, stream>>>(keys, stp);
  lsh_attn_kernel<<<B_ * CHUNKS, 256, 0, stream>>>(tgt, x, stp, tgt_mask, o4, sl4);
  combine_ln_kernel<<<B_ * S_, 256, 0, stream>>>(o4, sl4, lnw, lnb, out);

  // second output: y passthrough
  hipMemcpyAsync(out + BSD, y, BSD * sizeof(float), hipMemcpyDeviceToDevice,
                 stream);
}